// NET_40913858461758
// MI455X (gfx1250) — compile-verified
//
#include <hip/hip_runtime.h>
#include <math.h>

typedef float v2f __attribute__((ext_vector_type(2)));
typedef float v8f __attribute__((ext_vector_type(8)));

#define NN 50000
#define NE 800000
#define NH 8

// ---------------------------------------------------------------------------
// WMMA fp32 GEMM: C[M,N] = A[M,K] * B[K,N], row-major.
// One wave32 computes a 16x64 tile of C with V_WMMA_F32_16X16X4_F32.
// Requires M%16==0, N%64==0, K%4==0.
// A-frag layout (ISA 7.12.2, 32-bit A 16x4): lanes 0-15 -> K={k,k+1},
// lanes 16-31 -> K={k+2,k+3}; row = lane&15.
// B-frag: VGPR r holds row K=k+2*(lane>=16)+r, col = lane&15.
// C/D: VGPR v -> row v + 8*(lane>=16), col = lane&15.
// ---------------------------------------------------------------------------
__global__ __launch_bounds__(256) void gemm_wmma_f32(
    const float* __restrict__ A, const float* __restrict__ B,
    float* __restrict__ C, int M, int N, int K) {
  const int lane = threadIdx.x & 31;
  const int wave = (int)((blockIdx.x * blockDim.x + threadIdx.x) >> 5);
  const int nT = N >> 6;                    // 64-wide N tiles
  const int tm = wave / nT;
  const int tn = (wave % nT) << 6;
  if (tm >= (M >> 4)) return;
  const int l15 = lane & 15;
  const int kh  = (lane >> 4) << 1;         // 0 or 2: K-pair select per lane half
  const float* arow = A + (long)(tm * 16 + l15) * K + kh;
  const float* bcol = B + (long)kh * N + tn + l15;
  v8f c0 = {}, c1 = {}, c2 = {}, c3 = {};
  for (int k = 0; k < K; k += 4) {
    v2f a  = { arow[0],  arow[1] };
    v2f b0 = { bcol[0],  bcol[N] };
    v2f b1 = { bcol[16], bcol[16 + N] };
    v2f b2 = { bcol[32], bcol[32 + N] };
    v2f b3 = { bcol[48], bcol[48 + N] };
    c0 = __builtin_amdgcn_wmma_f32_16x16x4_f32(false, a, false, b0, (short)0, c0, false, false);
    c1 = __builtin_amdgcn_wmma_f32_16x16x4_f32(false, a, false, b1, (short)0, c1, false, false);
    c2 = __builtin_amdgcn_wmma_f32_16x16x4_f32(false, a, false, b2, (short)0, c2, false, false);
    c3 = __builtin_amdgcn_wmma_f32_16x16x4_f32(false, a, false, b3, (short)0, c3, false, false);
    arow += 4;
    bcol += (long)4 * N;
  }
  const int hi = (lane >> 4) << 3;
  float* crow = C + (long)(tm * 16 + hi) * N + tn + l15;
#pragma unroll
  for (int v = 0; v < 8; ++v) {
    crow[0]  = c0[v];
    crow[16] = c1[v];
    crow[32] = c2[v];
    crow[48] = c3[v];
    crow += N;
  }
}

// el[n,h] = feat[n,h,:].al[h,:] ; er[n,h] = feat[n,h,:].ar[h,:]
__global__ void attn_proj_kernel(const float* __restrict__ feat,
                                 const float* __restrict__ al,
                                 const float* __restrict__ ar,
                                 float* __restrict__ el, float* __restrict__ er,
                                 int total, int D) {
  int idx = blockIdx.x * blockDim.x + threadIdx.x;   // n*NH + h
  if (idx >= total) return;
  int h = idx & (NH - 1);
  const float* f  = feat + (long)idx * D;            // (n*NH+h)*D == n*NH*D + h*D
  const float* wl = al + h * D;
  const float* wr = ar + h * D;
  float sl = 0.f, sr = 0.f;
  for (int d = 0; d < D; ++d) { float v = f[d]; sl += v * wl[d]; sr += v * wr[d]; }
  el[idx] = sl;
  er[idx] = sr;
}

__global__ void fill_kernel(float* __restrict__ p, long n, float v) {
  long i = (long)blockIdx.x * blockDim.x + threadIdx.x;
  if (i < n) p[i] = v;
}

// Monotone-int-bits float atomic max (lattice op, order-independent).
__device__ __forceinline__ void atomic_max_f32(float* addr, float val) {
  if (val >= 0.f) atomicMax((int*)addr, __float_as_int(val));
  else            atomicMin((unsigned int*)addr, __float_as_uint(val));
}

// e = leaky_relu(el[src]+er[dst]); emax[dst] = segment max
__global__ void edge_logits_kernel(const float* __restrict__ el,
                                   const float* __restrict__ er,
                                   const int* __restrict__ src,
                                   const int* __restrict__ dst,
                                   float* __restrict__ e, float* __restrict__ emax) {
  long i = (long)blockIdx.x * blockDim.x + threadIdx.x;
  if (i >= (long)NE * NH) return;
  int ed = (int)(i >> 3);
  int h  = (int)(i & 7);
  int s = src[ed], t = dst[ed];
  float v = el[s * NH + h] + er[t * NH + h];
  v = (v >= 0.f) ? v : 0.2f * v;
  e[i] = v;
  atomic_max_f32(&emax[t * NH + h], v);
}

// where(isfinite(emax), emax, 0)
__global__ void emax_fix_kernel(float* __restrict__ emax, int n) {
  int i = blockIdx.x * blockDim.x + threadIdx.x;
  if (i >= n) return;
  float v = emax[i];
  emax[i] = (v >= -3.0e38f && v <= 3.0e38f) ? v : 0.f;
}

// ee = exp(e - emax[dst]); denom[dst] += ee   (ee stored in place of e)
__global__ void edge_exp_kernel(float* __restrict__ e,
                                const float* __restrict__ emax,
                                const int* __restrict__ dst,
                                float* __restrict__ denom) {
  long i = (long)blockIdx.x * blockDim.x + threadIdx.x;
  if (i >= (long)NE * NH) return;
  int ed = (int)(i >> 3);
  int h  = (int)(i & 7);
  int t = dst[ed];
  float ee = expf(e[i] - emax[t * NH + h]);
  e[i] = ee;
  atomicAdd(&denom[t * NH + h], ee);
}

// out[dst] += alpha * feat[src]  -- one wave32 per edge
__global__ __launch_bounds__(256) void edge_aggregate_kernel(
    const float* __restrict__ feat, const float* __restrict__ ee,
    const float* __restrict__ denom, const int* __restrict__ src,
    const int* __restrict__ dst, float* __restrict__ out, int D) {
  int lane = threadIdx.x & 31;
  long wave = ((long)blockIdx.x * blockDim.x + threadIdx.x) >> 5;
  if (wave >= NE) return;
  int s = src[wave], t = dst[wave];
  const float* frow = feat + (long)s * NH * D;
  float*       orow = out  + (long)t * NH * D;
  __builtin_prefetch(frow + lane * 8, 0, 1);        // global_prefetch over src row
  float myAlpha = 0.f;
  if (lane < NH)
    myAlpha = ee[wave * NH + lane] / (denom[t * NH + lane] + 1e-9f);
  int HD = NH * D;
  for (int i = lane; i < HD; i += 32) {
    float a = __shfl(myAlpha, i / D, 32);
    atomicAdd(&orow[i], a * frow[i]);
  }
}

__global__ void elu_kernel(float* __restrict__ p, long n) {
  long i = (long)blockIdx.x * blockDim.x + threadIdx.x;
  if (i >= n) return;
  float x = p[i];
  p[i] = (x > 0.f) ? x : expm1f(x);
}

// out[n,d] = mean_h h2[n,h,d]
__global__ void head_mean_kernel(const float* __restrict__ h2,
                                 float* __restrict__ out, int D) {
  long i = (long)blockIdx.x * blockDim.x + threadIdx.x;
  if (i >= (long)NN * D) return;
  long n = i / D;
  int  d = (int)(i % D);
  const float* row = h2 + n * NH * D + d;
  float s = 0.f;
#pragma unroll
  for (int h = 0; h < NH; ++h) s += row[h * D];
  out[i] = s * (1.0f / NH);
}

extern "C" void kernel_launch(void* const* d_in, const int* in_sizes, int n_in,
                              void* d_out, int out_size, void* d_ws, size_t ws_size,
                              hipStream_t stream) {
  (void)in_sizes; (void)n_in; (void)out_size; (void)ws_size;
  const float* X   = (const float*)d_in[0];
  const int*   src = (const int*)  d_in[1];
  const int*   dst = (const int*)  d_in[2];
  const float* W1  = (const float*)d_in[3];
  const float* al1 = (const float*)d_in[4];
  const float* ar1 = (const float*)d_in[5];
  const float* W2  = (const float*)d_in[6];
  const float* al2 = (const float*)d_in[7];
  const float* ar2 = (const float*)d_in[8];
  float* out = (float*)d_out;

  // Workspace layout (floats): 52.8M total = 211.2 MB
  float* ws   = (float*)d_ws;
  float* feat = ws;                  // 16,000,000 (feat1: 12.8M, then feat2: 16M)
  float* h1   = ws + 16000000L;      // 12,800,000
  float* h2   = ws + 28800000L;      // 16,000,000
  float* el   = ws + 44800000L;      //    400,000
  float* er   = ws + 45200000L;      //    400,000
  float* emax = ws + 45600000L;      //    400,000
  float* den  = ws + 46000000L;      //    400,000
  float* ebuf = ws + 46400000L;      //  6,400,000

  const int TPB = 256;
  const float NEG_INF = -__builtin_inff();
  const int gNH  = (NN * NH + TPB - 1) / TPB;          // 1563
  const int gEH  = (int)(((long)NE * NH + TPB - 1) / TPB); // 25000
  const int gEW  = (int)(((long)NE * 32 + TPB - 1) / TPB); // 100000

  // ================= Layer 1: H=8, D=32, 512 -> 256 =================
  {
    int waves = (NN / 16) * (256 / 64);                // 12500
    int blocks = (waves * 32 + TPB - 1) / TPB;
    gemm_wmma_f32<<<blocks, TPB, 0, stream>>>(X, W1, feat, NN, 256, 512);
  }
  attn_proj_kernel<<<gNH, TPB, 0, stream>>>(feat, al1, ar1, el, er, NN * NH, 32);
  fill_kernel<<<gNH, TPB, 0, stream>>>(emax, (long)NN * NH, NEG_INF);
  fill_kernel<<<gNH, TPB, 0, stream>>>(den,  (long)NN * NH, 0.f);
  fill_kernel<<<(int)((12800000L + TPB - 1) / TPB), TPB, 0, stream>>>(h1, 12800000L, 0.f);
  edge_logits_kernel<<<gEH, TPB, 0, stream>>>(el, er, src, dst, ebuf, emax);
  emax_fix_kernel<<<gNH, TPB, 0, stream>>>(emax, NN * NH);
  edge_exp_kernel<<<gEH, TPB, 0, stream>>>(ebuf, emax, dst, den);
  edge_aggregate_kernel<<<gEW, TPB, 0, stream>>>(feat, ebuf, den, src, dst, h1, 32);
  elu_kernel<<<(int)((12800000L + TPB - 1) / TPB), TPB, 0, stream>>>(h1, 12800000L);

  // ================= Layer 2: H=8, D=40, 256 -> 320 =================
  {
    int waves = (NN / 16) * (320 / 64);                // 15625
    int blocks = (waves * 32 + TPB - 1) / TPB;
    gemm_wmma_f32<<<blocks, TPB, 0, stream>>>(h1, W2, feat, NN, 320, 256);
  }
  attn_proj_kernel<<<gNH, TPB, 0, stream>>>(feat, al2, ar2, el, er, NN * NH, 40);
  fill_kernel<<<gNH, TPB, 0, stream>>>(emax, (long)NN * NH, NEG_INF);
  fill_kernel<<<gNH, TPB, 0, stream>>>(den,  (long)NN * NH, 0.f);
  fill_kernel<<<(int)((16000000L + TPB - 1) / TPB), TPB, 0, stream>>>(h2, 16000000L, 0.f);
  edge_logits_kernel<<<gEH, TPB, 0, stream>>>(el, er, src, dst, ebuf, emax);
  emax_fix_kernel<<<gNH, TPB, 0, stream>>>(emax, NN * NH);
  edge_exp_kernel<<<gEH, TPB, 0, stream>>>(ebuf, emax, dst, den);
  edge_aggregate_kernel<<<gEW, TPB, 0, stream>>>(feat, ebuf, den, src, dst, h2, 40);

  // Head mean -> out [50000, 40]
  head_mean_kernel<<<(int)(((long)NN * 40 + TPB - 1) / TPB), TPB, 0, stream>>>(h2, out, 40);
}